// Luong_exp_85040352461297
// MI455X (gfx1250) — compile-verified
//
#include <hip/hip_runtime.h>
#include <hip/hip_bf16.h>

#define BB 64
#define TT 256
#define EE 1024
#define UU 256
#define KX (EE + UU + UU)   // 1536 : [ctx | word | h] combined K for the gate GEMM
#define N4U (4 * UU)        // 1024 : i,f,g,o gates

typedef __attribute__((ext_vector_type(2))) float v2f;
typedef __attribute__((ext_vector_type(8))) float v8f;

__device__ __forceinline__ float sigmoidf_(float x) { return 1.0f / (1.0f + __expf(-x)); }

// ---------------------------------------------------------------------------
// Init: build Wcat = [kernel ; recurrent_kernel] (1536 x 1024) and copy h0/c0
// into workspace (inputs must not be mutated across graph replays).
// ---------------------------------------------------------------------------
__global__ void __launch_bounds__(256)
init_kernel(const float* __restrict__ Wk, const float* __restrict__ Wr,
            const float* __restrict__ h0, const float* __restrict__ c0,
            float* __restrict__ Wcat, float* __restrict__ h, float* __restrict__ c) {
    int i = blockIdx.x * blockDim.x + threadIdx.x;
    const int totalW = KX * N4U;
    if (i < totalW) {
        int k = i / N4U, n = i - k * N4U;
        Wcat[i] = (k < EE + UU) ? Wk[k * N4U + n] : Wr[(k - (EE + UU)) * N4U + n];
    }
    if (i < BB * UU) { h[i] = h0[i]; c[i] = c0[i]; }
}

// ---------------------------------------------------------------------------
// proj = enc (16384 x 1024) @ W_a (1024 x 256), fp32 WMMA 16x16x4.
// One wave per 16x16 C tile, K-loop of 256 WMMA steps. All operands L2-resident.
// A 16x4 f32 operand: lanes 0-15 -> M=lane, K={k0,k0+1}; lanes 16-31 -> K={k0+2,k0+3}.
// ---------------------------------------------------------------------------
__global__ void __launch_bounds__(256)
proj_gemm_kernel(const float* __restrict__ A, const float* __restrict__ Bw,
                 float* __restrict__ C) {
    const int lane = threadIdx.x & 31;
    const int wave = threadIdx.x >> 5;
    const int tile = blockIdx.x * (blockDim.x >> 5) + wave;   // 16384 tiles
    const int ntn  = UU / 16;                                  // 16 tiles along N
    const int m0 = (tile / ntn) * 16;
    const int n0 = (tile % ntn) * 16;

    const int half = lane >> 4;      // 0: K pair {0,1}, 1: K pair {2,3}
    const int l    = lane & 15;
    const int am   = m0 + l;
    const int bn   = n0 + l;
    const int kb   = half * 2;

    v8f acc = {};
    for (int k0 = 0; k0 < EE; k0 += 4) {
        v2f a, bv;
        a.x  = A[(size_t)am * EE + k0 + kb];
        a.y  = A[(size_t)am * EE + k0 + kb + 1];
        bv.x = Bw[(size_t)(k0 + kb) * UU + bn];
        bv.y = Bw[(size_t)(k0 + kb + 1) * UU + bn];
        acc = __builtin_amdgcn_wmma_f32_16x16x4_f32(false, a, false, bv,
                                                    (short)0, acc, false, false);
    }
#pragma unroll
    for (int i = 0; i < 8; ++i) {
        int m = m0 + i + 8 * half;   // C layout: vgpr i -> row i (+8 for hi half-wave)
        C[(size_t)m * UU + bn] = acc[i];
    }
}

// ---------------------------------------------------------------------------
// Per-step attention: scores = proj[b] @ h[b], softmax over T, ctx = alpha @ enc[b].
// One block per batch (64 blocks, 256 threads). Batched matvecs -> VALU + LDS
// reductions (no K-reuse, not a WMMA shape). Also assembles x = [ctx|word|h].
// ---------------------------------------------------------------------------
__global__ void __launch_bounds__(256)
attn_step_kernel(const float* __restrict__ proj, const float* __restrict__ enc,
                 const float* __restrict__ dec, const float* __restrict__ h,
                 float* __restrict__ x, float* __restrict__ attn_out,
                 float* __restrict__ attn_states, int step) {
    __shared__ float hs[UU];
    __shared__ float red[TT];
    __shared__ float al[TT];
    const int b   = blockIdx.x;
    const int tid = threadIdx.x;

    hs[tid] = h[b * UU + tid];
    __syncthreads();

    // scores[b][tid] = <proj[b][tid][:], h[b][:]>
    const float* pr = proj + ((size_t)b * TT + tid) * UU;
    float s = 0.f;
#pragma unroll 4
    for (int u = 0; u < UU; ++u) s += pr[u] * hs[u];

    red[tid] = s;
    __syncthreads();
    for (int off = 128; off > 0; off >>= 1) {
        if (tid < off) red[tid] = fmaxf(red[tid], red[tid + off]);
        __syncthreads();
    }
    float mx = red[0];
    __syncthreads();
    float ex = __expf(s - mx);
    red[tid] = ex;
    __syncthreads();
    for (int off = 128; off > 0; off >>= 1) {
        if (tid < off) red[tid] += red[tid + off];
        __syncthreads();
    }
    float alpha = ex / red[0];
    al[tid] = alpha;
    attn_states[((size_t)b * TT + tid) * TT + step] = alpha;  // (B, T_enc, T_step)
    __syncthreads();

    // ctx[b][e] = sum_t alpha[t] * enc[b][t][e]  (coalesced along e)
    const float* eb = enc + (size_t)b * TT * EE;
#pragma unroll
    for (int j = 0; j < 4; ++j) {
        int ei = tid + j * 256;
        float acc = 0.f;
        for (int t = 0; t < TT; ++t) acc += al[t] * eb[(size_t)t * EE + ei];
        attn_out[((size_t)b * TT + step) * EE + ei] = acc;
        x[b * KX + ei] = acc;
    }
    // x tail: decoder word for this step, then h
    x[b * KX + EE + tid]      = dec[((size_t)b * TT + step) * UU + tid];
    x[b * KX + EE + UU + tid] = hs[tid];
}

// ---------------------------------------------------------------------------
// Per-step fused LSTM gate GEMM: z = x (64x1536) @ Wcat (1536x1024) + bias,
// then gates. 16 blocks: block bu owns u-columns [16*bu, 16*bu+16) across ALL
// four gates and all 64 batch rows -> 16 WMMA tiles/block, 2 per wave (8 waves).
// Tiles land in LDS so each thread can fuse sigmoid/tanh and update h/c.
// ---------------------------------------------------------------------------
__global__ void __launch_bounds__(256)
lstm_step_kernel(const float* __restrict__ x, const float* __restrict__ Wcat,
                 const float* __restrict__ bias, float* __restrict__ h,
                 float* __restrict__ c) {
    __shared__ float zbuf[64 * 64];     // [m=0..63][gate*16 + uu]
    const int u0   = blockIdx.x * 16;
    const int lane = threadIdx.x & 31;
    const int wave = threadIdx.x >> 5;
    const int half = lane >> 4;
    const int l    = lane & 15;

#pragma unroll
    for (int sub = 0; sub < 2; ++sub) {
        const int tile = wave * 2 + sub;      // 0..15
        const int mt   = tile >> 2;           // row-tile 0..3 (batch 16-chunks)
        const int g    = tile & 3;            // gate: 0=i 1=f 2=g 3=o
        const int m0   = mt * 16;
        const int n0   = g * UU + u0;         // column in z (B x 4U)
        const int am   = m0 + l;
        const int bn   = n0 + l;
        const int kb   = half * 2;

        v8f acc = {};
        for (int k0 = 0; k0 < KX; k0 += 4) {
            v2f a, bv;
            a.x  = x[(size_t)am * KX + k0 + kb];
            a.y  = x[(size_t)am * KX + k0 + kb + 1];
            bv.x = Wcat[(size_t)(k0 + kb) * N4U + bn];
            bv.y = Wcat[(size_t)(k0 + kb + 1) * N4U + bn];
            acc = __builtin_amdgcn_wmma_f32_16x16x4_f32(false, a, false, bv,
                                                        (short)0, acc, false, false);
        }
        const float bval = bias[bn];
#pragma unroll
        for (int i = 0; i < 8; ++i) {
            int m = m0 + i + 8 * half;
            zbuf[m * 64 + g * 16 + l] = acc[i] + bval;
        }
    }
    __syncthreads();

    // 64 batches x 16 u's = 1024 (b,u) pairs; 4 per thread.
#pragma unroll
    for (int j = 0; j < 4; ++j) {
        int p  = threadIdx.x + j * 256;
        int m  = p >> 4;        // batch
        int uu = p & 15;
        float iv = zbuf[m * 64 + 0  + uu];
        float fv = zbuf[m * 64 + 16 + uu];
        float gv = zbuf[m * 64 + 32 + uu];
        float ov = zbuf[m * 64 + 48 + uu];
        int idx  = m * UU + u0 + uu;
        float cn = sigmoidf_(fv) * c[idx] + sigmoidf_(iv) * tanhf(gv);
        float hn = sigmoidf_(ov) * tanhf(cn);
        c[idx] = cn;
        h[idx] = hn;
    }
}

// ---------------------------------------------------------------------------
extern "C" void kernel_launch(void* const* d_in, const int* in_sizes, int n_in,
                              void* d_out, int out_size, void* d_ws, size_t ws_size,
                              hipStream_t stream) {
    const float* enc  = (const float*)d_in[0];   // (B,T,E)
    const float* dec  = (const float*)d_in[1];   // (B,T,U)
    const float* h0   = (const float*)d_in[2];   // (B,U)
    const float* c0   = (const float*)d_in[3];   // (B,U)
    const float* Wa   = (const float*)d_in[4];   // (E,U)
    const float* Wk   = (const float*)d_in[5];   // (E+U, 4U)
    const float* Wr   = (const float*)d_in[6];   // (U, 4U)
    const float* bias = (const float*)d_in[7];   // (4U)

    float* attn_out    = (float*)d_out;                          // (B,T,E)
    float* attn_states = attn_out + (size_t)BB * TT * EE;        // (B,T,T)

    float* ws   = (float*)d_ws;
    float* proj = ws;                               // B*T*U      (16.8 MB)
    float* Wcat = proj + (size_t)BB * TT * UU;      // KX*4U      ( 6.3 MB)
    float* xbuf = Wcat + (size_t)KX * N4U;          // B*KX
    float* hbuf = xbuf + (size_t)BB * KX;           // B*U
    float* cbuf = hbuf + (size_t)BB * UU;           // B*U

    const int initN = KX * N4U;
    init_kernel<<<(initN + 255) / 256, 256, 0, stream>>>(Wk, Wr, h0, c0, Wcat, hbuf, cbuf);

    const int tiles = (BB * TT / 16) * (UU / 16);   // 16384 wave-tiles
    proj_gemm_kernel<<<tiles / 8, 256, 0, stream>>>(enc, Wa, proj);

    for (int step = 0; step < TT; ++step) {
        attn_step_kernel<<<BB, 256, 0, stream>>>(proj, enc, dec, hbuf, xbuf,
                                                 attn_out, attn_states, step);
        lstm_step_kernel<<<UU / 16, 256, 0, stream>>>(xbuf, Wcat, bias, hbuf, cbuf);
    }
}